// MyModel_87454124081955
// MI455X (gfx1250) — compile-verified
//
#include <hip/hip_runtime.h>

typedef __attribute__((ext_vector_type(2))) float v2f;
typedef __attribute__((ext_vector_type(8))) float v8f;

#define T_STEPS 28
#define IN_DIM  28
#define HID     8
#define NCLS    10
#define WAVES_PER_BLOCK 4

__device__ __forceinline__ float sig_f(float x) {
    // sigmoid, robust at extremes: exp(-x)=inf -> 0 ; exp(-x)=0 -> 1
    return 1.0f / (1.0f + __expf(-x));
}
__device__ __forceinline__ float tanh_f(float x) {
    // tanh via exp, robust at extremes: exp(2x)=inf -> 1 ; exp(2x)=0 -> -1
    return 1.0f - 2.0f / (1.0f + __expf(2.0f * x));
}

__global__ __launch_bounds__(WAVES_PER_BLOCK * 32)
void lstm_wmma_f32_kernel(const float* __restrict__ x,   // [B][28][28]
                          const float* __restrict__ Wl,  // [36][32] row-major, gates i,j,f,o
                          const float* __restrict__ bl,  // [32]
                          const float* __restrict__ ow,  // [8][10]
                          const float* __restrict__ ob,  // [10]
                          float* __restrict__ out,       // [B][10]
                          int Btot)
{
    __shared__ __align__(16) float gbuf[WAVES_PER_BLOCK][16 * 32]; // gates, [row][gate]
    __shared__ __align__(16) float hbuf[WAVES_PER_BLOCK][16 * HID]; // h state, [row][h]

    const int lane = threadIdx.x & 31;
    const int wv   = threadIdx.x >> 5;
    const int tileBase = (blockIdx.x * WAVES_PER_BLOCK + wv) * 16;

    const int rowA = lane & 15;              // A/C row (M) served by this lane's half
    const int koff = (lane < 16) ? 0 : 2;    // K sub-offset within a 4-wide chunk
    const int colN = lane & 15;              // N column served by this lane

    // ---- W (B-matrix) fragments: 9 K-chunks x 2 N-tiles, resident in VGPRs ----
    // B 4x16 layout: VGPR0 = K{0|2}, VGPR1 = K{1|3} by lane half -> same koff as A.
    v2f wB[9][2];
    #pragma unroll
    for (int c = 0; c < 9; ++c) {
        const int k0 = 4 * c + koff;
        #pragma unroll
        for (int tdx = 0; tdx < 2; ++tdx) {
            const int col = tdx * 16 + colN;
            v2f b;
            b.x = Wl[(k0 + 0) * 32 + col];
            b.y = Wl[(k0 + 1) * 32 + col];
            wB[c][tdx] = b;
        }
    }
    const float biasLo = bl[colN];        // gates 0..15  (i, j)
    const float biasHi = bl[16 + colN];   // gates 16..31 (f, o)

    // per-lane cell state: 4 fixed (row,h) pairs; p = lane*4+q -> row=p>>3, h=p&7
    float cst[4] = {0.f, 0.f, 0.f, 0.f};
    #pragma unroll
    for (int q = 0; q < 4; ++q) {
        const int p = lane * 4 + q;
        hbuf[wv][(p >> 3) * HID + (p & 7)] = 0.0f;   // h0 = 0
    }
    __builtin_amdgcn_wave_barrier();

    // clamp row for address safety when B is not a multiple of 16
    int grow = tileBase + rowA;
    if (grow > Btot - 1) grow = Btot - 1;
    const float* xrow = x + (size_t)grow * T_STEPS * IN_DIM;

    for (int t = 0; t < T_STEPS; ++t) {
        const float* xt = xrow + t * IN_DIM;
        if (t + 1 < T_STEPS)
            __builtin_prefetch(xt + IN_DIM, 0, 0);   // next timestep's row slice

        // accumulators preloaded with bias (broadcast along rows)
        v8f acc0, acc1;
        #pragma unroll
        for (int v = 0; v < 8; ++v) { acc0[v] = biasLo; acc1[v] = biasHi; }

        // K chunks 0..6: from x_t (global, aligned float2 per lane)
        #pragma unroll
        for (int c = 0; c < 7; ++c) {
            v2f a = *(const v2f*)(xt + 4 * c + koff);
            acc0 = __builtin_amdgcn_wmma_f32_16x16x4_f32(false, a, false, wB[c][0],
                                                         (short)0, acc0, false, false);
            acc1 = __builtin_amdgcn_wmma_f32_16x16x4_f32(false, a, false, wB[c][1],
                                                         (short)0, acc1, false, false);
        }
        // K chunks 7..8: from h (LDS, already in A-fragment layout)
        #pragma unroll
        for (int c = 7; c < 9; ++c) {
            v2f a = *(const v2f*)(&hbuf[wv][rowA * HID + (c - 7) * 4 + koff]);
            acc0 = __builtin_amdgcn_wmma_f32_16x16x4_f32(false, a, false, wB[c][0],
                                                         (short)0, acc0, false, false);
            acc1 = __builtin_amdgcn_wmma_f32_16x16x4_f32(false, a, false, wB[c][1],
                                                         (short)0, acc1, false, false);
        }

        // C layout -> LDS [row][gate]
        #pragma unroll
        for (int v = 0; v < 8; ++v) {
            const int row = v + ((lane < 16) ? 0 : 8);
            gbuf[wv][row * 32 + colN]      = acc0[v];
            gbuf[wv][row * 32 + 16 + colN] = acc1[v];
        }
        __builtin_amdgcn_wave_barrier();   // intra-wave DS in-order; pin scheduling

        // LSTM cell update: i,j,f,o ; new_c = c*sig(f+1) + sig(i)*tanh(j)
        #pragma unroll
        for (int q = 0; q < 4; ++q) {
            const int p = lane * 4 + q;
            const int row = p >> 3, h = p & 7;
            const float* g = &gbuf[wv][row * 32];
            const float gi = g[h], gj = g[8 + h], gf = g[16 + h], go = g[24 + h];
            const float nc = cst[q] * sig_f(gf + 1.0f) + sig_f(gi) * tanh_f(gj);
            cst[q] = nc;
            hbuf[wv][row * HID + h] = tanh_f(nc) * sig_f(go);   // A-layout for next step
        }
        __builtin_amdgcn_wave_barrier();
    }

    // logits = h_T @ ow + ob : 16x10 = 160 outputs, 5 per lane
    #pragma unroll
    for (int q = 0; q < 5; ++q) {
        const int p = lane * 5 + q;           // 0..159
        const int row = p / 10, cls = p % 10;
        float s = ob[cls];
        #pragma unroll
        for (int h = 0; h < HID; ++h)
            s += hbuf[wv][row * HID + h] * ow[h * 10 + cls];
        const int gr = tileBase + row;
        if (gr < Btot) out[(size_t)gr * NCLS + cls] = s;
    }
}

extern "C" void kernel_launch(void* const* d_in, const int* in_sizes, int n_in,
                              void* d_out, int out_size, void* d_ws, size_t ws_size,
                              hipStream_t stream) {
    const float* x  = (const float*)d_in[0];
    const float* Wl = (const float*)d_in[1];
    const float* bl = (const float*)d_in[2];
    const float* ow = (const float*)d_in[3];
    const float* ob = (const float*)d_in[4];
    float* out = (float*)d_out;

    const int Btot  = in_sizes[0] / (T_STEPS * IN_DIM);   // 32768
    const int tiles = (Btot + 15) / 16;                   // 16 batch rows per wave
    const int blocks = (tiles + WAVES_PER_BLOCK - 1) / WAVES_PER_BLOCK;

    lstm_wmma_f32_kernel<<<blocks, WAVES_PER_BLOCK * 32, 0, stream>>>(
        x, Wl, bl, ow, ob, out, Btot);
}